// OcclusionSecondLayer_2482491097129
// MI455X (gfx1250) — compile-verified
//
#include <hip/hip_runtime.h>
#include <hip/hip_bf16.h>

typedef __attribute__((ext_vector_type(2))) float v2f;
typedef __attribute__((ext_vector_type(8))) float v8f;

constexpr int SIZE_IN  = 16384;   // K
constexpr int SIZE_OUT = 8192;    // M
constexpr int KCHUNKS  = 8;
constexpr int KCHUNK   = SIZE_IN / KCHUNKS;   // 2048

// Accumulate a 16-row tile of W times x over [kbase, kbase+KLEN) using
// V_WMMA_F32_16X16X4_F32.  B broadcasts x across all 16 N columns, so every
// column of the 16x16 accumulator equals the GEMV result for this tile.
//
// f32 16x4 A layout (ISA 7.12.2): lanes 0-15 -> rows 0-15, K=0 (V0), K=1 (V1);
// lanes 16-31 -> rows 0-15, K=2 (V0), K=3 (V1).  B (4x16) splits K the same way
// across lane halves, so with x broadcast over N, lane L needs x[k + 2*(L>>4)]
// and x[k + 2*(L>>4) + 1] -> one b64 load, identical addressing to A.
//
// `tile`/`kbase` must be wave-uniform (callers pass readfirstlane'd values) so
// the backend can keep the W/x bases in SGPRs (GVS addressing: saddr + 32-bit
// voffset + immediate ioffset) instead of per-lane 64-bit address arithmetic.
template <int KLEN>
__device__ inline v8f tile_gemv(const float* __restrict__ W,
                                const float* __restrict__ x,
                                int tile, int kbase, int lane) {
  const int m = lane & 15;
  const int h = lane >> 4;                 // 0 or 1
  // Wave-uniform bases; per-lane 32-bit element offsets (max 1 MB for W tile).
  const float* wbase = W + (size_t)tile * 16 * SIZE_IN + kbase;
  const float* xbase = x + kbase;
  const int woff = m * SIZE_IN + 2 * h;    // per-lane, fits easily in i32
  const int xoff = 2 * h;

  v8f c = {};
#pragma unroll 16
  for (int k = 0; k < KLEN; k += 4) {
    // W: 512 MB streamed exactly once -> non-temporal, don't pollute L2.
    v2f a = __builtin_nontemporal_load((const v2f*)(wbase + woff + k));
    // x: 64 KB reused by all 4096 waves -> regular temporal (L2-resident).
    v2f b = *(const v2f*)(xbase + xoff + k);
    // (neg_a, A, neg_b, B, c_mod, C, reuse_a, reuse_b)
    c = __builtin_amdgcn_wmma_f32_16x16x4_f32(false, a, false, b,
                                              (short)0, c, false, false);
  }
  return c;
}

// C/D layout: lanes 0-15, VGPR j -> M=j (any N); lanes 16-31, VGPR j -> M=8+j.
// All columns identical, so lane 0 writes rows 0-7, lane 16 writes rows 8-15.
__device__ inline void store_tile(float* __restrict__ dst, const v8f& c,
                                  int lane, const float* __restrict__ bias) {
  if (lane == 0) {
#pragma unroll
    for (int j = 0; j < 8; ++j)
      dst[j] = c[j] + (bias ? bias[j] : 0.0f);
  } else if (lane == 16) {
#pragma unroll
    for (int j = 0; j < 8; ++j)
      dst[8 + j] = c[j] + (bias ? bias[8 + j] : 0.0f);
  }
}

// Phase 1: 512 tiles x 8 K-chunks = 4096 waves streaming W at full bandwidth.
__global__ void __launch_bounds__(256)
gemv_wmma_partial(const float* __restrict__ W, const float* __restrict__ x,
                  float* __restrict__ partial) {
  const int lane = threadIdx.x & 31;
  const int gid  = blockIdx.x * (blockDim.x >> 5) + (threadIdx.x >> 5);
  // Wave-uniform by construction; make it provable for the backend.
  const int tile  = __builtin_amdgcn_readfirstlane(gid / KCHUNKS);   // 0..511
  const int chunk = __builtin_amdgcn_readfirstlane(gid % KCHUNKS);   // 0..7
  v8f c = tile_gemv<KCHUNK>(W, x, tile, chunk * KCHUNK, lane);
  store_tile(partial + (size_t)chunk * SIZE_OUT + tile * 16, c, lane, nullptr);
}

// Phase 2: fold the 8 partials and add bias (deterministic reduction).
__global__ void __launch_bounds__(256)
reduce_bias(const float* __restrict__ partial, const float* __restrict__ bias,
            float* __restrict__ out) {
  const int i = blockIdx.x * blockDim.x + threadIdx.x;
  if (i >= SIZE_OUT) return;
  float s = bias[i];
#pragma unroll
  for (int cchunk = 0; cchunk < KCHUNKS; ++cchunk)
    s += partial[(size_t)cchunk * SIZE_OUT + i];
  out[i] = s;
}

// Fallback if the workspace is too small: one wave per tile, full K sweep,
// bias fused.  (512 waves; less latency hiding but needs no scratch.)
__global__ void __launch_bounds__(256)
gemv_wmma_full(const float* __restrict__ W, const float* __restrict__ x,
               const float* __restrict__ bias, float* __restrict__ out) {
  const int lane = threadIdx.x & 31;
  const int tile = __builtin_amdgcn_readfirstlane(
      blockIdx.x * (blockDim.x >> 5) + (threadIdx.x >> 5));
  v8f c = tile_gemv<SIZE_IN>(W, x, tile, 0, lane);
  store_tile(out + tile * 16, c, lane, bias + tile * 16);
}

extern "C" void kernel_launch(void* const* d_in, const int* in_sizes, int n_in,
                              void* d_out, int out_size, void* d_ws, size_t ws_size,
                              hipStream_t stream) {
  const float* x    = (const float*)d_in[0];   // 16384
  const float* W    = (const float*)d_in[1];   // 8192 x 16384, row-major
  const float* bias = (const float*)d_in[2];   // 8192
  float* out = (float*)d_out;

  const size_t need = (size_t)KCHUNKS * SIZE_OUT * sizeof(float);  // 256 KB
  if (ws_size >= need) {
    const int total_waves = (SIZE_OUT / 16) * KCHUNKS;   // 4096
    const int waves_per_block = 8;                        // 256 threads
    gemv_wmma_partial<<<total_waves / waves_per_block, 256, 0, stream>>>(
        W, x, (float*)d_ws);
    reduce_bias<<<(SIZE_OUT + 255) / 256, 256, 0, stream>>>(
        (const float*)d_ws, bias, out);
  } else {
    const int total_waves = SIZE_OUT / 16;                // 512
    gemv_wmma_full<<<total_waves / 8, 256, 0, stream>>>(W, x, bias, out);
  }
}